// Aggregator_46858093199903
// MI455X (gfx1250) — compile-verified
//
#include <hip/hip_runtime.h>

typedef __attribute__((ext_vector_type(2))) float v2f;
typedef __attribute__((ext_vector_type(8))) float v8f;

#define N_NODES   100000
#define N_EDGES   1600000
#define NODE_DIM  32
#define EDGE_DIM  16
#define D_IN      48      // NODE_DIM + EDGE_DIM
#define OUT_DIM   32
#define W_PITCH   49      // LDS row pitch (gcd(49,64)==1 -> conflict-free)

// ---------------- kernel 0: zero the scatter workspace ----------------
__global__ void zero_ws_kernel(float* __restrict__ ws, int n) {
  int i = blockIdx.x * blockDim.x + threadIdx.x;
  int stride = gridDim.x * blockDim.x;
  for (; i < n; i += stride) ws[i] = 0.0f;
}

// ---------------- kernel 1: edge messages + atomic scatter ----------------
// m = ReLU(concat(y[src], ex) @ W1^T + b1);  accum[dst] += m; cnt[dst] += 1
__global__ __launch_bounds__(256) void edge_kernel(
    const float* __restrict__ y, const float* __restrict__ ex,
    const float* __restrict__ W1, const float* __restrict__ b1,
    const int* __restrict__ src, const int* __restrict__ dst,
    float* __restrict__ accum,   // [N_NODES * 48]
    float* __restrict__ cnt)     // [N_NODES]
{
  __shared__ float ldsW1[D_IN * W_PITCH];
  __shared__ float ldsB1[D_IN];

  const int tid = threadIdx.x;
  for (int i = tid; i < D_IN * D_IN; i += 256)
    ldsW1[(i / D_IN) * W_PITCH + (i % D_IN)] = W1[i];
  if (tid < D_IN) ldsB1[tid] = b1[tid];
  __syncthreads();

  const int lane = tid & 31;
  const int wave = tid >> 5;
  const int row  = lane & 15;   // M row of A frag / N col of B & C frags
  const int hi   = lane >> 4;   // half-wave select
  const int koff = hi * 2;      // A: lanes 0-15 hold K={0,1}, 16-31 hold K={2,3}

  const int nTiles = N_EDGES / 16;   // 100000, exact
  for (int tile = blockIdx.x * 8 + wave; tile < nTiles; tile += gridDim.x * 8) {
    const int e  = tile * 16 + row;
    const int si = src[e];

    // ---- A fragments: feat[row][4s+koff .. +1], s = 0..11 ----
    v2f a[12];
    const float* yp = y + (size_t)si * NODE_DIM;
    #pragma unroll
    for (int s = 0; s < 8; ++s)
      a[s] = *(const v2f*)(yp + 4 * s + koff);
    const float* ep = ex + (size_t)e * EDGE_DIM;
    #pragma unroll
    for (int s = 8; s < 12; ++s)
      a[s] = *(const v2f*)(ep + 4 * (s - 8) + koff);

    // ---- C init with bias (per-column broadcast: col = t*16 + row) ----
    v8f acc[3];
    #pragma unroll
    for (int t = 0; t < 3; ++t) {
      const float bv = ldsB1[t * 16 + row];
      #pragma unroll
      for (int j = 0; j < 8; ++j) acc[t][j] = bv;
    }

    // ---- 12 K-steps x 3 N-tiles of V_WMMA_F32_16X16X4_F32 ----
    // B frag: B[k][n] = W1[n][k]; lane holds n = row, k = 4s+koff..+1
    #pragma unroll
    for (int s = 0; s < 12; ++s) {
      #pragma unroll
      for (int t = 0; t < 3; ++t) {
        v2f b = *(const v2f*)(&ldsW1[(t * 16 + row) * W_PITCH + 4 * s + koff]);
        acc[t] = __builtin_amdgcn_wmma_f32_16x16x4_f32(
            false, a[s], false, b, (short)0, acc[t], false, false);
      }
    }

    // ---- dst indices for my 8 C rows (M = j + hi*8) ----
    int dvals[8];
    #pragma unroll
    for (int j = 0; j < 8; ++j) dvals[j] = dst[tile * 16 + hi * 8 + j];

    // ---- ReLU + scatter: C layout lane->col(t*16+row), VGPR j->row(j+hi*8) ----
    #pragma unroll
    for (int t = 0; t < 3; ++t) {
      #pragma unroll
      for (int j = 0; j < 8; ++j) {
        float v = acc[t][j];
        v = v > 0.0f ? v : 0.0f;
        unsafeAtomicAdd(&accum[(size_t)dvals[j] * D_IN + t * 16 + row], v);
      }
    }
    if (hi == 0) unsafeAtomicAdd(&cnt[dst[e]], 1.0f);  // one edge per low lane
  }
}

// ---------------- kernel 2: mean + node update ----------------
// z = accum / max(cnt,1) (0 if isolated);  h = ReLU(z @ W2^T + b2)
__global__ __launch_bounds__(256) void node_kernel(
    const float* __restrict__ accum, const float* __restrict__ cnt,
    const float* __restrict__ W2, const float* __restrict__ b2,
    float* __restrict__ h)       // [N_NODES * 32]
{
  __shared__ float ldsW2[OUT_DIM * W_PITCH];
  __shared__ float ldsB2[OUT_DIM];

  const int tid = threadIdx.x;
  for (int i = tid; i < OUT_DIM * D_IN; i += 256)
    ldsW2[(i / D_IN) * W_PITCH + (i % D_IN)] = W2[i];
  if (tid < OUT_DIM) ldsB2[tid] = b2[tid];
  __syncthreads();

  const int lane = tid & 31;
  const int wave = tid >> 5;
  const int row  = lane & 15;
  const int hi   = lane >> 4;
  const int koff = hi * 2;

  const int nTiles = N_NODES / 16;   // 6250, exact
  for (int tile = blockIdx.x * 8 + wave; tile < nTiles; tile += gridDim.x * 8) {
    const int node = tile * 16 + row;
    const float c = cnt[node];
    const float r = c > 0.0f ? 1.0f / c : 0.0f;   // isolated nodes -> z = 0

    v2f a[12];
    const float* zp = accum + (size_t)node * D_IN;
    #pragma unroll
    for (int s = 0; s < 12; ++s) {
      v2f v = *(const v2f*)(zp + 4 * s + koff);
      a[s][0] = v[0] * r;
      a[s][1] = v[1] * r;
    }

    v8f acc[2];
    #pragma unroll
    for (int t = 0; t < 2; ++t) {
      const float bv = ldsB2[t * 16 + row];
      #pragma unroll
      for (int j = 0; j < 8; ++j) acc[t][j] = bv;
    }

    #pragma unroll
    for (int s = 0; s < 12; ++s) {
      #pragma unroll
      for (int t = 0; t < 2; ++t) {
        v2f b = *(const v2f*)(&ldsW2[(t * 16 + row) * W_PITCH + 4 * s + koff]);
        acc[t] = __builtin_amdgcn_wmma_f32_16x16x4_f32(
            false, a[s], false, b, (short)0, acc[t], false, false);
      }
    }

    #pragma unroll
    for (int t = 0; t < 2; ++t) {
      #pragma unroll
      for (int j = 0; j < 8; ++j) {
        float v = acc[t][j];
        v = v > 0.0f ? v : 0.0f;
        h[(size_t)(tile * 16 + hi * 8 + j) * OUT_DIM + t * 16 + row] = v;
      }
    }
  }
}

// ---------------- launch ----------------
extern "C" void kernel_launch(void* const* d_in, const int* in_sizes, int n_in,
                              void* d_out, int out_size, void* d_ws, size_t ws_size,
                              hipStream_t stream) {
  const float* y  = (const float*)d_in[0];
  const float* ex = (const float*)d_in[1];
  const float* W1 = (const float*)d_in[2];
  const float* b1 = (const float*)d_in[3];
  const float* W2 = (const float*)d_in[4];
  const float* b2 = (const float*)d_in[5];
  const int* src  = (const int*)d_in[6];
  const int* dst  = (const int*)d_in[7];
  float* out = (float*)d_out;

  // workspace: accum [N_NODES*48] floats, then cnt [N_NODES] floats (19.6 MB)
  float* accum = (float*)d_ws;
  float* cnt   = accum + (size_t)N_NODES * D_IN;

  const int zeroN = N_NODES * (D_IN + 1);
  zero_ws_kernel<<<2048, 256, 0, stream>>>(accum, zeroN);

  // 100000 edge tiles, 8 waves/block -> exactly one tile per wave
  edge_kernel<<<12500, 256, 0, stream>>>(y, ex, W1, b1, src, dst, accum, cnt);

  // 6250 node tiles
  node_kernel<<<782, 256, 0, stream>>>(accum, cnt, W2, b2, out);
}